// MultiElementWiseAffine_15736760172656
// MI455X (gfx1250) — compile-verified
//
#include <hip/hip_runtime.h>
#include <hip/hip_bf16.h>
#include <stdint.h>

// MultiElementWiseAffine for MI455X (gfx1250).
//
// out[i,k] = fma(x_i, dm[t_i,k], cm[t_i,k]) with fused, pre-masked tables
//   dm[t,k] = (k < len[t]) ? disc[t,k] : 0
//   cm[t,k] = (k < len[t]) ? disc[t,k]*off[t,k] : 0
//
// Store-bandwidth bound: 256MB of output -> ~11.5us floor at 23.3 TB/s.
// CDNA5 paths used (all verified in gfx1250 disassembly):
//   - global_load_async_to_lds_b128 + s_wait_asynccnt (ASYNCcnt DMA of tables)
//   - ds_load_b128 from LDS fused tables
//   - non-temporal global_store_b128 (output > 192MB L2)
//   - global_prefetch_b8 (WGP scope) for the input/task streams

#define N_TASKS 16
#define LMAX    64
#define THREADS 256
#define ITERS   16   // float4s per thread per block => 256 rows / block

typedef float f4 __attribute__((ext_vector_type(4)));

__global__ __launch_bounds__(THREADS)
void mewa_kernel(const float* __restrict__ input,
                 const int*   __restrict__ task_ids,
                 const float* __restrict__ offsets,
                 const float* __restrict__ disc,
                 const int*   __restrict__ lengths,
                 float* __restrict__ out)
{
    __shared__ __align__(16) float raw_off [N_TASKS * LMAX]; // 4KB
    __shared__ __align__(16) float raw_disc[N_TASKS * LMAX]; // 4KB
    __shared__ __align__(16) float dm      [N_TASKS * LMAX]; // 4KB fused disc (masked)
    __shared__ __align__(16) float cm      [N_TASKS * LMAX]; // 4KB fused disc*off (masked)
    __shared__ int s_len[N_TASKS];

    const unsigned tid = threadIdx.x;

    // ---- Stage 1: async DMA the 8KB of tables into LDS (ASYNCcnt path). ----
    // 256 threads x 16B each covers the full 4096B of each table.
    {
        unsigned goff = tid * 16u;  // byte offset into the [16x64] f32 tables
        // Truncating a flat pointer to a __shared__ object to 32 bits yields
        // the LDS byte offset (flat LDS aperture = {SHARED_BASE, offset}).
        unsigned lds_o = (unsigned)(uint64_t)(uintptr_t)&raw_off [tid * 4u];
        unsigned lds_d = (unsigned)(uint64_t)(uintptr_t)&raw_disc[tid * 4u];
        asm volatile("global_load_async_to_lds_b128 %0, %1, %2"
                     :: "v"(lds_o), "v"(goff), "s"(offsets) : "memory");
        asm volatile("global_load_async_to_lds_b128 %0, %1, %2"
                     :: "v"(lds_d), "v"(goff), "s"(disc) : "memory");
    }
    if (tid < N_TASKS) s_len[tid] = lengths[tid];

    // Prefetch this block's input / task_id rows. Locality 3 -> WGP-scope
    // prefetch: fills all cache levels (SYS-scope would stop at GL2, but the
    // consumers are this block's lanes a few hundred cycles later).
    const unsigned row_base = blockIdx.x * (THREADS * ITERS / 16u); // 256 rows/block
    __builtin_prefetch(input    + row_base + tid, 0, 3);
    __builtin_prefetch(task_ids + row_base + tid, 0, 3);

    asm volatile("s_wait_asynccnt 0" ::: "memory");
    __syncthreads();

    // ---- Stage 2: build fused, pre-masked tables (removes all branching). ----
#pragma unroll
    for (int e = tid; e < N_TASKS * LMAX; e += THREADS) {
        int   t = e >> 6;
        int   k = e & 63;
        float d = raw_disc[e];
        float o = raw_off[e];
        bool  v = k < s_len[t];
        dm[e] = v ? d : 0.0f;
        cm[e] = v ? d * o : 0.0f;
    }
    __syncthreads();

    // ---- Stage 3: stream the output. One float4 (b128) per lane per iter. ----
    // Global float4 index space: B*16. Consecutive lanes -> consecutive float4
    // stores (fully coalesced b128). 16 lanes share one row's x/t (same-address
    // loads coalesce to a single request, so read traffic stays ~8MB).
    f4* __restrict__ out4 = (f4*)out;
#pragma unroll
    for (int it = 0; it < ITERS; ++it) {
        unsigned g   = blockIdx.x * (THREADS * ITERS) + it * THREADS + tid;
        unsigned row = g >> 4;          // output row
        unsigned k0  = (g & 15u) << 2;  // starting column (multiple of 4)

        float x = input[row];
        int   t = task_ids[row];

        const f4 d = *(const f4*)&dm[(t << 6) + k0];  // ds_load_b128
        const f4 c = *(const f4*)&cm[(t << 6) + k0];  // ds_load_b128

        f4 o;
        o.x = __builtin_fmaf(x, d.x, c.x);
        o.y = __builtin_fmaf(x, d.y, c.y);
        o.z = __builtin_fmaf(x, d.z, c.z);
        o.w = __builtin_fmaf(x, d.w, c.w);

        // Output (256MB) exceeds L2 (192MB): non-temporal store keeps L2 for
        // the input/task/table reads.
        __builtin_nontemporal_store(o, &out4[g]);
    }
}

extern "C" void kernel_launch(void* const* d_in, const int* in_sizes, int n_in,
                              void* d_out, int out_size, void* d_ws, size_t ws_size,
                              hipStream_t stream)
{
    const float* input    = (const float*)d_in[0];   // [B,1] f32
    const int*   task_ids = (const int*)  d_in[1];   // [B]   i32
    const float* offsets  = (const float*)d_in[2];   // [16,64] f32
    const float* disc     = (const float*)d_in[3];   // [16,64] f32
    const int*   lengths  = (const int*)  d_in[4];   // [16]  i32
    float*       out      = (float*)d_out;           // [B,64] f32

    const int B = in_sizes[0];                       // 1048576
    const int rows_per_block = THREADS * ITERS / 16; // 256
    const int grid = B / rows_per_block;             // 4096

    mewa_kernel<<<grid, THREADS, 0, stream>>>(input, task_ids, offsets, disc,
                                              lengths, out);
}